// FirstDeriv_52561809768586
// MI455X (gfx1250) — compile-verified
//
#include <hip/hip_runtime.h>
#include <hip/hip_bf16.h>

// CDNA5 / gfx1250 — FirstDeriv batched weighted least-squares gradient.
// Strategy: fp32 end-to-end (weights span ~1e-0..1e8, low precision is unsafe).
// Gram accumulation G = sum_k w * m m^T (m = [dv.x dv.y dv.z du]) is done on
// the matrix pipe with V_WMMA_F32_16X16X4_F32, packing 4 nodes block-diagonally
// into the 16x16 tile (rows/cols = (node,component)), contracting 4 neighbors
// per WMMA. Gathers of coords/y are L2-resident (3.2 MB << 192 MB L2).

typedef __attribute__((ext_vector_type(2))) float v2f;
typedef __attribute__((ext_vector_type(8))) float v8f;

#define KNBR   64
#define EPS_W  1e-8f
#define LAM    1e-6f

__global__ __launch_bounds__(256) void firstderiv_wmma_kernel(
    const float* __restrict__ coords,   // [N,3]
    const int*   __restrict__ nbr,      // [N,K]
    const float* __restrict__ yv,       // [N,1]
    float*       __restrict__ out,      // [N,3]
    int n, int kdim)
{
    // per-wave staging: [wave][node g][neighbor kk][component]
    __shared__ float s_wm[8][4][8][4];   // w * m
    __shared__ float s_mm[8][4][8][4];   // m
    __shared__ float s_D [8][16][16];    // 16x16 WMMA accumulator dump

    const int lane = threadIdx.x & 31;
    const int wave = threadIdx.x >> 5;

    // ---- prep-phase mapping: lane -> (node g, neighbor-in-iter kk) ----
    const int g  = lane >> 3;        // 0..3
    const int kk = lane & 7;         // 0..7

    // ---- fragment mapping: lane -> (half, row/col rc -> (fg, fi)) ----
    const int half = lane >> 4;      // 0: K={0,1}, 1: K={2,3}
    const int rc   = lane & 15;      // row (A) == col (B) index in 16
    const int fg   = rc >> 2;        // node of this row/col
    const int fi   = rc & 3;         // component of this row/col

    long long base = ((long long)blockIdx.x * 8 + wave) * 4;  // first node of wave
    long long npll = base + g;
    if (npll > (long long)(n - 1)) npll = (long long)(n - 1); // clamp: uniform flow
    const int node_p = (int)npll;

    // center point for this prep lane's node (invariant over the k loop)
    const float cx = coords[node_p * 3 + 0];
    const float cy = coords[node_p * 3 + 1];
    const float cz = coords[node_p * 3 + 2];
    const float cu = yv[node_p];

    v8f acc = {};  // 16x16 fp32 accumulator (4 diagonal 4x4 Gram blocks useful)

    const int iters = kdim >> 3;     // 8 neighbors per iteration
    for (int it = 0; it < iters; ++it) {
        const int k = it * 8 + kk;
        const int nb = nbr[(long long)node_p * kdim + k];

        float m0 = 0.f, m1 = 0.f, m2 = 0.f, m3 = 0.f;
        if (nb >= 0) {
            m0 = coords[nb * 3 + 0] - cx;
            m1 = coords[nb * 3 + 1] - cy;
            m2 = coords[nb * 3 + 2] - cz;
            m3 = yv[nb] - cu;
        }
        const float w = 1.0f / (m0 * m0 + m1 * m1 + m2 * m2 + EPS_W);

        s_wm[wave][g][kk][0] = w * m0;
        s_wm[wave][g][kk][1] = w * m1;
        s_wm[wave][g][kk][2] = w * m2;
        s_wm[wave][g][kk][3] = w * m3;
        s_mm[wave][g][kk][0] = m0;
        s_mm[wave][g][kk][1] = m1;
        s_mm[wave][g][kk][2] = m2;
        s_mm[wave][g][kk][3] = m3;
        __syncthreads();

        // two K=4 chunks per iteration
        #pragma unroll
        for (int c = 0; c < 2; ++c) {
            const int k0 = c * 4 + 2 * half;   // this half's K pair base
            v2f a, b;
            // A (16x4 fp32): VGPR0 = K {0|2}, VGPR1 = K {1|3}; row M = (fg,fi)
            a.x = s_wm[wave][fg][k0 + 0][fi];
            a.y = s_wm[wave][fg][k0 + 1][fi];
            // B (4x16 fp32): VGPR0 = row K {0|2}, VGPR1 = row K {1|3}; col N = (fg,fi)
            b.x = s_mm[wave][fg][k0 + 0][fi];
            b.y = s_mm[wave][fg][k0 + 1][fi];
            acc = __builtin_amdgcn_wmma_f32_16x16x4_f32(
                /*neg_a=*/false, a, /*neg_b=*/false, b,
                /*c_mod=*/(short)0, acc, /*reuse_a=*/false, /*reuse_b=*/false);
        }
        __syncthreads();   // protect s_wm/s_mm before next iteration overwrites
    }

    // dump D to LDS: VGPR r, lane L -> D[r + 8*half][L&15]
    #pragma unroll
    for (int r = 0; r < 8; ++r)
        s_D[wave][r + 8 * half][rc] = acc[r];
    __syncthreads();

    // ---- per-node 3x3 regularized solve (Cramer), lanes 0..3 ----
    if (lane < 4) {
        const long long node = base + lane;
        if (node < (long long)n) {
            const int o = 4 * lane;
            const float a00 = s_D[wave][o + 0][o + 0] + LAM;
            const float a01 = s_D[wave][o + 0][o + 1];
            const float a02 = s_D[wave][o + 0][o + 2];
            const float a10 = s_D[wave][o + 1][o + 0];
            const float a11 = s_D[wave][o + 1][o + 1] + LAM;
            const float a12 = s_D[wave][o + 1][o + 2];
            const float a20 = s_D[wave][o + 2][o + 0];
            const float a21 = s_D[wave][o + 2][o + 1];
            const float a22 = s_D[wave][o + 2][o + 2] + LAM;
            const float b0  = s_D[wave][o + 0][o + 3];
            const float b1  = s_D[wave][o + 1][o + 3];
            const float b2  = s_D[wave][o + 2][o + 3];

            const float det =
                a00 * (a11 * a22 - a12 * a21)
              - a01 * (a10 * a22 - a12 * a20)
              + a02 * (a10 * a21 - a11 * a20);
            const float inv_det = 1.0f / det;

            const float x0 = (b0  * (a11 * a22 - a12 * a21)
                            - a01 * (b1  * a22 - a12 * b2 )
                            + a02 * (b1  * a21 - a11 * b2 )) * inv_det;
            const float x1 = (a00 * (b1  * a22 - a12 * b2 )
                            - b0  * (a10 * a22 - a12 * a20)
                            + a02 * (a10 * b2  - b1  * a20)) * inv_det;
            const float x2 = (a00 * (a11 * b2  - b1  * a21)
                            - a01 * (a10 * b2  - b1  * a20)
                            + b0  * (a10 * a21 - a11 * a20)) * inv_det;

            float* op = out + (long long)node * 3;
            op[0] = x0;
            op[1] = x1;
            op[2] = x2;
        }
    }
}

extern "C" void kernel_launch(void* const* d_in, const int* in_sizes, int n_in,
                              void* d_out, int out_size, void* d_ws, size_t ws_size,
                              hipStream_t stream) {
    const float* coords = (const float*)d_in[0];   // [N,3] fp32
    const int*   nbr    = (const int*)d_in[1];     // [N,K] int
    const float* yv     = (const float*)d_in[2];   // [N,1] fp32
    float*       out    = (float*)d_out;           // [N,3] fp32

    const int n    = in_sizes[0] / 3;
    const int kdim = (n > 0) ? (in_sizes[1] / n) : KNBR;

    // 8 waves/block * 4 nodes/wave = 32 nodes per block
    const int blocks = (n + 31) / 32;
    firstderiv_wmma_kernel<<<blocks, 256, 0, stream>>>(coords, nbr, yv, out, n, kdim);
}